// GanaGAT_27522150433357
// MI455X (gfx1250) — compile-verified
//
#include <hip/hip_runtime.h>
#include <hip/hip_bf16.h>
#include <math.h>

#define NN    50000
#define EE    1600000
#define ETOT  (EE + NN)
#define KK    256      // inner dim of every GEMM (F_IN and 8*32)
#define HC    256      // HEADS*HID
#define H0    8        // heads layer 0
#define C0    32       // per-head channels layer 0
#define NCLS  40

typedef __attribute__((ext_vector_type(2))) float v2f;
typedef __attribute__((ext_vector_type(8))) float v8f;

// ---------- order-preserving float<->int encoding for atomic max ----------
__device__ __forceinline__ int enc_f32(float x) {
  int u = __float_as_int(x);
  return u >= 0 ? u : (u ^ 0x7fffffff);
}
__device__ __forceinline__ float dec_f32(int k) {
  return __int_as_float(k >= 0 ? k : (k ^ 0x7fffffff));
}

// ---------- fills ----------
__global__ void fill_f32(float* __restrict__ p, float v, int n) {
  int i = blockIdx.x * blockDim.x + threadIdx.x;
  if (i < n) p[i] = v;
}
__global__ void fill_i32(int* __restrict__ p, int v, int n) {
  int i = blockIdx.x * blockDim.x + threadIdx.x;
  if (i < n) p[i] = v;
}

// ---------- XP[M,NCOLS] = Xin[M,256] @ W[256,NCOLS] via fp32 WMMA 16x16x4 ----------
// Block = 256 threads = 8 waves. Block owns one 16-wide column tile (nt) shared
// via LDS (transposed, padded); each wave computes one 16x16 C tile (mt = bm*8+wave).
template <int NCOLS>
__global__ void gemm_wmma_f32(const float* __restrict__ Xin, const float* __restrict__ W,
                              float* __restrict__ XP, int M) {
  constexpr int NTN = (NCOLS + 15) / 16;
  constexpr int KP  = KK + 2;                 // pad 2 floats -> conflict-free ds reads
  __shared__ float Bt[16][KP];                // W tile, transposed: Bt[col][k]

  const int bn = blockIdx.x % NTN;
  const int bm = blockIdx.x / NTN;
  const int n0 = bn * 16;

  // cooperative, coalesced load of W[:, n0:n0+16] into LDS (transposed)
  for (int i = threadIdx.x; i < 16 * KK; i += blockDim.x) {
    int kb = i >> 4;
    int cc = i & 15;
    int col = n0 + cc;
    Bt[cc][kb] = (col < NCOLS) ? W[(size_t)kb * NCOLS + col] : 0.0f;
  }
  __syncthreads();

  const int wave = threadIdx.x >> 5;
  const int lane = threadIdx.x & 31;
  const int mt = bm * 8 + wave;
  if (mt * 16 >= M) return;

  const int half = lane >> 4;                 // 0: K=k..k+1, 1: K=k+2..k+3
  const int l    = lane & 15;
  const int m0   = mt * 16;
  const float* arow = Xin + (size_t)(m0 + l) * KK + 2 * half;   // 8B aligned
  const float* brow = &Bt[l][2 * half];                          // 8B aligned

  v8f c = {};
  for (int k = 0; k < KK; k += 4) {
    v2f a = *(const v2f*)(arow + k);          // global_load_b64
    v2f b = *(const v2f*)(brow + k);          // ds_load_b64
    c = __builtin_amdgcn_wmma_f32_16x16x4_f32(false, a, false, b, (short)0, c, false, false);
  }

  const int col = n0 + l;
  if (col < NCOLS) {
    for (int v = 0; v < 8; ++v)
      XP[(size_t)(m0 + v + 8 * half) * NCOLS + col] = c[v];
  }
}

// ---------- per-node attention dots: asrc[n,h]=<xp[n,h,:],as[h,:]> ----------
template <int H, int C>
__global__ void att_dots(const float* __restrict__ xp, const float* __restrict__ as,
                         const float* __restrict__ ad, float* __restrict__ asrc,
                         float* __restrict__ adst, int Nn) {
  int t = blockIdx.x * blockDim.x + threadIdx.x;
  if (t >= Nn * H) return;
  int h = t % H;
  int n = t / H;
  const float* row = xp + (size_t)n * H * C + (size_t)h * C;
  float s = 0.f, d = 0.f;
  for (int c = 0; c < C; ++c) {
    float v = row[c];
    s += v * as[h * C + c];
    d += v * ad[h * C + c];
  }
  asrc[t] = s;
  adst[t] = d;
}

// ---------- edge pass 1: segment max of leaky_relu(asrc[src]+adst[dst]) ----------
template <int H>
__global__ void edge_max(const int* __restrict__ es, const int* __restrict__ ed,
                         const float* __restrict__ asrc, const float* __restrict__ adst,
                         int* __restrict__ mx) {
  int t = blockIdx.x * blockDim.x + threadIdx.x;
  if (t >= ETOT * H) return;
  int e = t / H, h = t - e * H;
  int s = e < EE ? es[e] : (e - EE);
  int d = e < EE ? ed[e] : (e - EE);
  float x = asrc[s * H + h] + adst[d * H + h];
  x = x > 0.f ? x : 0.2f * x;
  atomicMax(&mx[d * H + h], enc_f32(x));
}

// ---------- edge pass 2: denom[d,h] += exp(e - m[d,h]) ----------
template <int H>
__global__ void edge_expsum(const int* __restrict__ es, const int* __restrict__ ed,
                            const float* __restrict__ asrc, const float* __restrict__ adst,
                            const int* __restrict__ mx, float* __restrict__ den) {
  int t = blockIdx.x * blockDim.x + threadIdx.x;
  if (t >= ETOT * H) return;
  int e = t / H, h = t - e * H;
  int s = e < EE ? es[e] : (e - EE);
  int d = e < EE ? ed[e] : (e - EE);
  float x = asrc[s * H + h] + adst[d * H + h];
  x = x > 0.f ? x : 0.2f * x;
  atomicAdd(&den[d * H + h], expf(x - dec_f32(mx[d * H + h])));
}

// ---------- edge pass 3: out[d,:] += alpha * xp[s,:]  (wave per edge; H*C==256) ----------
template <int H, int C>
__global__ void edge_msg(const int* __restrict__ es, const int* __restrict__ ed,
                         const float* __restrict__ asrc, const float* __restrict__ adst,
                         const int* __restrict__ mx, const float* __restrict__ den,
                         const float* __restrict__ xp, float* __restrict__ out) {
  constexpr int HCl = H * C;                 // 256
  int wave = (int)((blockIdx.x * blockDim.x + threadIdx.x) >> 5);
  int lane = threadIdx.x & 31;
  if (wave >= ETOT) return;
  int s = wave < EE ? es[wave] : (wave - EE);
  int d = wave < EE ? ed[wave] : (wave - EE);
  int c0 = lane * (HCl / 32);                // 8 contiguous channels, inside one head
  int h = c0 / C;
  float x = asrc[s * H + h] + adst[d * H + h];
  x = x > 0.f ? x : 0.2f * x;
  float alpha = expf(x - dec_f32(mx[d * H + h])) / den[d * H + h];
  const float4* xr = (const float4*)(xp + (size_t)s * HCl + c0);  // 16B aligned
  float4 v0 = xr[0];                          // global_load_b128, coalesced
  float4 v1 = xr[1];
  float* orow = out + (size_t)d * HCl + c0;
  atomicAdd(&orow[0], v0.x * alpha);
  atomicAdd(&orow[1], v0.y * alpha);
  atomicAdd(&orow[2], v0.z * alpha);
  atomicAdd(&orow[3], v0.w * alpha);
  atomicAdd(&orow[4], v1.x * alpha);
  atomicAdd(&orow[5], v1.y * alpha);
  atomicAdd(&orow[6], v1.z * alpha);
  atomicAdd(&orow[7], v1.w * alpha);
}

// ---------- edge pass 3 (narrow last layer): thread per (edge, channel), H==1 ----------
__global__ void edge_msg_narrow(const int* __restrict__ es, const int* __restrict__ ed,
                                const float* __restrict__ asrc, const float* __restrict__ adst,
                                const int* __restrict__ mx, const float* __restrict__ den,
                                const float* __restrict__ xp, float* __restrict__ out, int C) {
  long long t = (long long)blockIdx.x * blockDim.x + threadIdx.x;
  if (t >= (long long)ETOT * C) return;
  int e = (int)(t / C);
  int c = (int)(t - (long long)e * C);
  int s = e < EE ? es[e] : (e - EE);
  int d = e < EE ? ed[e] : (e - EE);
  float x = asrc[s] + adst[d];
  x = x > 0.f ? x : 0.2f * x;
  float alpha = expf(x - dec_f32(mx[d])) / den[d];
  atomicAdd(&out[(size_t)d * C + c], xp[(size_t)s * C + c] * alpha);
}

// ---------- bias + optional ELU ----------
__global__ void bias_elu(const float* __restrict__ in, const float* __restrict__ b,
                         float* __restrict__ out, int Nn, int W, int do_elu) {
  int t = blockIdx.x * blockDim.x + threadIdx.x;
  if (t >= Nn * W) return;
  float v = in[t] + b[t % W];
  if (do_elu) v = v > 0.f ? v : expm1f(v);
  out[t] = v;
}

// ---------- final: log_softmax(in + b) per row of C ----------
__global__ void logsoftmax_rows(const float* __restrict__ in, const float* __restrict__ b,
                                float* __restrict__ out, int Nn, int C) {
  int n = blockIdx.x * blockDim.x + threadIdx.x;
  if (n >= Nn) return;
  const float* r = in + (size_t)n * C;
  float mxv = -3.4e38f;
  for (int c = 0; c < C; ++c) {
    float v = r[c] + b[c];
    mxv = v > mxv ? v : mxv;
  }
  float se = 0.f;
  for (int c = 0; c < C; ++c) se += expf(r[c] + b[c] - mxv);
  float lse = logf(se);
  for (int c = 0; c < C; ++c) out[(size_t)n * C + c] = r[c] + b[c] - mxv - lse;
}

static inline int nblk(long long threads, int bs) { return (int)((threads + bs - 1) / bs); }

extern "C" void kernel_launch(void* const* d_in, const int* in_sizes, int n_in,
                              void* d_out, int out_size, void* d_ws, size_t ws_size,
                              hipStream_t stream) {
  const float* x   = (const float*)d_in[0];
  const int*   ei  = (const int*)  d_in[1];
  const float* W0  = (const float*)d_in[2];
  const float* as0 = (const float*)d_in[3];
  const float* ad0 = (const float*)d_in[4];
  const float* b0  = (const float*)d_in[5];
  const float* W1  = (const float*)d_in[6];
  const float* as1 = (const float*)d_in[7];
  const float* ad1 = (const float*)d_in[8];
  const float* b1  = (const float*)d_in[9];
  const float* W2  = (const float*)d_in[10];
  const float* as2 = (const float*)d_in[11];
  const float* ad2 = (const float*)d_in[12];
  const float* b2  = (const float*)d_in[13];
  const int* esrc = ei;
  const int* edst = ei + EE;
  float* out = (float*)d_out;

  // workspace carve-up (floats)
  float* A    = (float*)d_ws;                       // N*256
  float* B    = A    + (size_t)NN * HC;             // N*256
  float* XP2  = B    + (size_t)NN * HC;             // N*40
  float* OUT2 = XP2  + (size_t)NN * NCLS;           // N*40
  float* ASRC = OUT2 + (size_t)NN * NCLS;           // N*8
  float* ADST = ASRC + (size_t)NN * H0;             // N*8
  int*   MX   = (int*)(ADST + (size_t)NN * H0);     // N*8
  float* DEN  = (float*)(MX + (size_t)NN * H0);     // N*8

  const int BS = 256;
  const int INT_MIN_V = (int)0x80000000;
  const int MT = NN / 16;                // 3125 row tiles
  const int BM = (MT + 7) / 8;           // 391 row-blocks (8 waves each)

  // ================= Layer 0: 256 -> 8 heads x 32, concat =================
  {
    gemm_wmma_f32<HC><<<BM * (HC / 16), BS, 0, stream>>>(x, W0, A, NN);
    att_dots<H0, C0><<<nblk((long long)NN * H0, BS), BS, 0, stream>>>(A, as0, ad0, ASRC, ADST, NN);
    fill_i32<<<nblk((long long)NN * H0, BS), BS, 0, stream>>>(MX, INT_MIN_V, NN * H0);
    fill_f32<<<nblk((long long)NN * H0, BS), BS, 0, stream>>>(DEN, 0.f, NN * H0);
    fill_f32<<<nblk((long long)NN * HC, BS), BS, 0, stream>>>(B, 0.f, NN * HC);
    edge_max<H0><<<nblk((long long)ETOT * H0, BS), BS, 0, stream>>>(esrc, edst, ASRC, ADST, MX);
    edge_expsum<H0><<<nblk((long long)ETOT * H0, BS), BS, 0, stream>>>(esrc, edst, ASRC, ADST, MX, DEN);
    edge_msg<H0, C0><<<nblk((long long)ETOT * 32, BS), BS, 0, stream>>>(esrc, edst, ASRC, ADST, MX, DEN, A, B);
    bias_elu<<<nblk((long long)NN * HC, BS), BS, 0, stream>>>(B, b0, A, NN, HC, 1);  // h1 -> A
  }

  // ================= Layer 1: 256 -> 256, 1 head =================
  {
    gemm_wmma_f32<HC><<<BM * (HC / 16), BS, 0, stream>>>(A, W1, B, NN);
    att_dots<1, HC><<<nblk((long long)NN, BS), BS, 0, stream>>>(B, as1, ad1, ASRC, ADST, NN);
    fill_i32<<<nblk((long long)NN, BS), BS, 0, stream>>>(MX, INT_MIN_V, NN);
    fill_f32<<<nblk((long long)NN, BS), BS, 0, stream>>>(DEN, 0.f, NN);
    fill_f32<<<nblk((long long)NN * HC, BS), BS, 0, stream>>>(A, 0.f, NN * HC);
    edge_max<1><<<nblk((long long)ETOT, BS), BS, 0, stream>>>(esrc, edst, ASRC, ADST, MX);
    edge_expsum<1><<<nblk((long long)ETOT, BS), BS, 0, stream>>>(esrc, edst, ASRC, ADST, MX, DEN);
    edge_msg<1, HC><<<nblk((long long)ETOT * 32, BS), BS, 0, stream>>>(esrc, edst, ASRC, ADST, MX, DEN, B, A);
    bias_elu<<<nblk((long long)NN * HC, BS), BS, 0, stream>>>(A, b1, A, NN, HC, 1);  // h2 -> A (in place)
  }

  // ================= Layer 2: 256 -> 40, 1 head, no concat (mean==identity) ====
  {
    gemm_wmma_f32<NCLS><<<BM * ((NCLS + 15) / 16), BS, 0, stream>>>(A, W2, XP2, NN);
    att_dots<1, NCLS><<<nblk((long long)NN, BS), BS, 0, stream>>>(XP2, as2, ad2, ASRC, ADST, NN);
    fill_i32<<<nblk((long long)NN, BS), BS, 0, stream>>>(MX, INT_MIN_V, NN);
    fill_f32<<<nblk((long long)NN, BS), BS, 0, stream>>>(DEN, 0.f, NN);
    fill_f32<<<nblk((long long)NN * NCLS, BS), BS, 0, stream>>>(OUT2, 0.f, NN * NCLS);
    edge_max<1><<<nblk((long long)ETOT, BS), BS, 0, stream>>>(esrc, edst, ASRC, ADST, MX);
    edge_expsum<1><<<nblk((long long)ETOT, BS), BS, 0, stream>>>(esrc, edst, ASRC, ADST, MX, DEN);
    edge_msg_narrow<<<nblk((long long)ETOT * NCLS, BS), BS, 0, stream>>>(esrc, edst, ASRC, ADST, MX, DEN, XP2, OUT2, NCLS);
    logsoftmax_rows<<<nblk((long long)NN, BS), BS, 0, stream>>>(OUT2, b2, out, NN, NCLS);
  }
}